// EwaldPotential_54314156425687
// MI455X (gfx1250) — compile-verified
//
#include <hip/hip_runtime.h>

// CDNA5 / gfx1250 wave32 WMMA types
typedef __attribute__((ext_vector_type(2))) float v2f;  // A/B for 16x16x4 f32 (2 VGPRs)
typedef __attribute__((ext_vector_type(8))) float v8f;  // C/D 16x16 f32 (8 VGPRs)

#define INV_4PI    0.07957747154594767f   // 1/(2*pi)/2
#define SELF_COEF  0.06349363593424098f   // 1/(sigma * (2*pi)^1.5), sigma=1
#define INV_SQRT2  0.7071067811865476f    // 1/(sigma*sqrt(2))
#define EPS_R      1e-6f

// Branchless erf(x) for x >= 0: Abramowitz & Stegun 7.1.26, |err| <= 1.5e-7.
// Straight-line: 1 rcp + 5 fma + 1 native exp + 1 fma. No EXEC changes.
__device__ __forceinline__ float erf_pos(float x) {
    const float t = __builtin_amdgcn_rcpf(fmaf(0.3275911f, x, 1.0f));
    float p = fmaf(1.061405429f, t, -1.453152027f);
    p = fmaf(p, t, 1.421413741f);
    p = fmaf(p, t, -0.284496736f);
    p = fmaf(p, t, 0.254829592f);
    const float e = __expf(-x * x);          // native v_exp_f32 path
    return fmaf(-(p * t), e, 1.0f);
}

__global__ void ewald_zero_kernel(float* out) {
    if (blockIdx.x == 0 && threadIdx.x == 0) out[0] = 0.0f;
}

// norms[i] = |pos_i|^2
__global__ void ewald_norms_kernel(const float* __restrict__ pos,
                                   float* __restrict__ norms, int N) {
    int i = blockIdx.x * blockDim.x + threadIdx.x;
    if (i < N) {
        const float* p = pos + (size_t)i * 3;
        norms[i] = fmaf(p[0], p[0], fmaf(p[1], p[1], p[2] * p[2]));
    }
}

// Gaussian self-interaction: sum(q*q) * SELF_COEF
__global__ void ewald_self_kernel(const float* __restrict__ q, float* out, int total) {
    __shared__ float red[8];
    float v = 0.0f;
    for (int i = blockIdx.x * blockDim.x + threadIdx.x; i < total;
         i += gridDim.x * blockDim.x) {
        float x = q[i];
        v = fmaf(x, x, v);
    }
    for (int off = 16; off > 0; off >>= 1) v += __shfl_xor(v, off, 32);
    const int lane = threadIdx.x & 31;
    const int w    = threadIdx.x >> 5;
    if (lane == 0) red[w] = v;
    __syncthreads();
    if (threadIdx.x == 0) {
        float s = 0.0f;
        const int nw = blockDim.x >> 5;
        for (int i = 0; i < nw; ++i) s += red[i];
        atomicAdd(out, s * SELF_COEF);
    }
}

// Pairwise erf-screened sum. One wave per 16x16 tile; tj >= ti, off-diag x2.
// USE_NORMS resolved at compile time -> no scalar branching in the hot loop.
template <bool USE_NORMS>
__global__ void ewald_pair_kernel(const float* __restrict__ pos,
                                  const float* __restrict__ q,
                                  const float* __restrict__ norms,
                                  float* out, int N) {
    const int      lane   = threadIdx.x & 31;
    const int      half   = lane >> 4;      // 0: K/M-low half, 1: K/M-high half
    const int      m16    = lane & 15;      // row (A) / col (B,C,D) index
    const unsigned wave   = (blockIdx.x * blockDim.x + threadIdx.x) >> 5;
    const unsigned nWaves = (gridDim.x * blockDim.x) >> 5;
    const unsigned Nt     = (unsigned)N >> 4;            // #16-tiles per dim
    const unsigned T      = Nt * (Nt + 1u) / 2u;         // lower-tri tile count

    float acc = 0.0f;

    for (unsigned t = wave; t < T; t += nWaves) {
        // t -> (a,b), a >= b : a = col tile, b = row tile (uniform per wave)
        unsigned a = (unsigned)((sqrtf(8.0f * (float)t + 1.0f) - 1.0f) * 0.5f);
        while (a * (a + 1u) / 2u > t) --a;
        while ((a + 1u) * (a + 2u) / 2u <= t) ++a;
        const unsigned b  = t - a * (a + 1u) / 2u;
        const unsigned i0 = b << 4;          // row block base
        const unsigned j0 = a << 4;          // col block base
        const float    wgt = (a == b) ? 1.0f : 2.0f;

        // ---- Load WMMA operands (all loads unconditional; selects are cndmask) ----
        // A (16x4, MxK): lane m16 = M; halves hold K={0,1} / K={2,3}
        // B (4x16, KxN): lane m16 = N; same K striping (B = q[j]^T / pos[j]^T)
        v2f aq, bq, ar, br;
        {
            const float* qa = q + (size_t)(i0 + m16) * 4;
            aq.x = qa[2 * half];
            aq.y = qa[2 * half + 1];
            const float* qb = q + (size_t)(j0 + m16) * 4;
            bq.x = qb[2 * half];
            bq.y = qb[2 * half + 1];

            const float* pa = pos + (size_t)(i0 + m16) * 3;
            const float pax = pa[0], pay = pa[1], paz = pa[2];
            ar.x = half ? paz : pax;          // K=0 / K=2
            ar.y = half ? 0.0f : pay;         // K=1 / K=3 (zero pad)

            const float* pb = pos + (size_t)(j0 + m16) * 3;
            const float pbx = pb[0], pby = pb[1], pbz = pb[2];
            br.x = half ? pbz : pbx;
            br.y = half ? 0.0f : pby;
        }

        v8f zero = {};
        // S[i,j] = sum_c q[i,c] q[j,c]
        v8f S = __builtin_amdgcn_wmma_f32_16x16x4_f32(
            false, aq, false, bq, (short)0, zero, false, false);
        // G[i,j] = r_i . r_j
        v8f G = __builtin_amdgcn_wmma_f32_16x16x4_f32(
            false, ar, false, br, (short)0, zero, false, false);

        // |r|^2 terms. C/D layout: VGPR v holds M = v + 8*half, N = m16.
        float nj;
        float niv[8];
        if (USE_NORMS) {
            nj = norms[j0 + m16];
            // rows i0+8*half .. i0+8*half+7 are contiguous & 32B aligned
            const float4 nlo = *(const float4*)(norms + i0 + 8 * half);
            const float4 nhi = *(const float4*)(norms + i0 + 8 * half + 4);
            niv[0] = nlo.x; niv[1] = nlo.y; niv[2] = nlo.z; niv[3] = nlo.w;
            niv[4] = nhi.x; niv[5] = nhi.y; niv[6] = nhi.z; niv[7] = nhi.w;
        } else {
            const float* pj = pos + (size_t)(j0 + m16) * 3;
            nj = fmaf(pj[0], pj[0], fmaf(pj[1], pj[1], pj[2] * pj[2]));
#pragma unroll
            for (int v = 0; v < 8; ++v) {
                const float* pi = pos + (size_t)(i0 + v + 8 * half) * 3;
                niv[v] = fmaf(pi[0], pi[0], fmaf(pi[1], pi[1], pi[2] * pi[2]));
            }
        }

        float tileSum = 0.0f;
#pragma unroll
        for (int v = 0; v < 8; ++v) {
            // d2 = ni + nj - 2*G ; clamp tiny negative fp error
            const float d2 = fmaxf(fmaf(-2.0f, G[v], niv[v] + nj), 0.0f);
            const float r  = __builtin_amdgcn_sqrtf(d2);
            // erf(0)*rcp(eps) == 0 -> diagonal naturally contributes nothing
            const float kern = erf_pos(r * INV_SQRT2) *
                               __builtin_amdgcn_rcpf(r + EPS_R);
            tileSum = fmaf(S[v], kern, tileSum);
        }
        acc = fmaf(wgt, tileSum, acc);
    }

    // wave32 reduction
    for (int off = 16; off > 0; off >>= 1) acc += __shfl_xor(acc, off, 32);
    __shared__ float red[8];
    const int w = threadIdx.x >> 5;
    if (lane == 0) red[w] = acc;
    __syncthreads();
    if (threadIdx.x == 0) {
        float s = 0.0f;
        const int nw = blockDim.x >> 5;
        for (int i = 0; i < nw; ++i) s += red[i];
        atomicAdd(out, s * INV_4PI);
    }
}

extern "C" void kernel_launch(void* const* d_in, const int* in_sizes, int n_in,
                              void* d_out, int out_size, void* d_ws, size_t ws_size,
                              hipStream_t stream) {
    const float* pos = (const float*)d_in[0];   // [N,3] f32
    const float* q   = (const float*)d_in[1];   // [N,C] f32, C==4
    float* out = (float*)d_out;                  // [1] f32

    const int N  = in_sizes[0] / 3;              // 8192
    const int NC = in_sizes[1];                  // N*C

    const bool useNorms = (ws_size >= (size_t)N * sizeof(float));
    float* norms = useNorms ? (float*)d_ws : nullptr;

    // serialize on stream: zero -> self -> (norms) -> pairs; self and pairs atomicAdd into out
    ewald_zero_kernel<<<1, 1, 0, stream>>>(out);
    ewald_self_kernel<<<32, 256, 0, stream>>>(q, out, NC);
    // 1024 blocks x 8 waves = 8192 waves over ~131k tiles (~16 tiles/wave)
    if (useNorms) {
        ewald_norms_kernel<<<(N + 255) / 256, 256, 0, stream>>>(pos, norms, N);
        ewald_pair_kernel<true><<<1024, 256, 0, stream>>>(pos, q, norms, out, N);
    } else {
        ewald_pair_kernel<false><<<1024, 256, 0, stream>>>(pos, q, nullptr, out, N);
    }
}